// NGP_INTERP_ENC_987842478250
// MI455X (gfx1250) — compile-verified
//
#include <hip/hip_runtime.h>
#include <cstdint>

// NGP hash-grid interpolation encoding for MI455X (gfx1250, wave32).
// Gather/L2-bound: B*L*8 = 33.5M random 8-byte table gathers, 64MB table
// resident in the 192MB L2; HBM only sees the cold fill + 32MB NT output.
// WMMA is not applicable (batched 1x8 @ 8x2 contraction on gathered
// per-lane data). Optimization levers: one b64 gather per corner, uniform
// SGPR base + 32-bit scale_offset addressing (level folded into the index),
// minimal hash ALU, coalesced wave32 NT stores.

namespace {

constexpr int      kLevels = 16;
constexpr uint32_t kTBits  = 19;
constexpr uint32_t kTMask  = (1u << kTBits) - 1u;
constexpr uint32_t kPrime1 = 2654435761u;
constexpr uint32_t kPrime2 = 805459861u;

typedef float v2f __attribute__((ext_vector_type(2)));

// floor(16 * b^l), b = exp(ln(512/16)/15) = 2^(1/3)  (matches numpy RES)
__constant__ float kResTab[kLevels] = {
    16.f,  20.f,  25.f,  32.f,  40.f,  50.f,  64.f,  80.f,
    101.f, 128.f, 161.f, 203.f, 256.f, 322.f, 406.f, 512.f};

__global__ __launch_bounds__(256) void ngp_interp_enc_kernel(
    const float* __restrict__ x,       // (B, 3)
    const float* __restrict__ tables,  // (L, 2^19, 2) = 2^23 float2, 64MB
    float* __restrict__ out,           // (B, L*2)
    int npts) {
  const uint32_t tid = blockIdx.x * 256u + threadIdx.x;
  const uint32_t b   = tid >> 4;    // point index
  const uint32_t l   = tid & 15u;   // level index (lane-minor -> coalesced out)
  if (b >= (uint32_t)npts) return;

  // 16 lanes share one point: same-address broadcast b96 load.
  const float* xp = x + (size_t)b * 3u;
  const float x0 = xp[0];
  const float x1 = xp[1];
  const float x2 = xp[2];

  const float res = kResTab[l];
  const float s0 = x0 * res;
  const float s1 = x1 * res;
  const float s2 = x2 * res;

  const float f0 = floorf(s0);
  const float f1 = floorf(s1);
  const float f2 = floorf(s2);
  const uint32_t g0 = (uint32_t)f0;
  const uint32_t g1 = (uint32_t)f1;
  const uint32_t g2 = (uint32_t)f2;

  // Hash components per dim for corner offsets {0,1}: (g+c)*P = g*P (+P)
  // -> only 2 v_mul_lo_u32 total.
  // (a^b^c)&M == (a&M)^(b&M)^(c&M): mask components up front.
  //  - x components g0, g0+1 are < 2^19 already (grid <= 512): no mask.
  //  - fold the level's table offset (l<<19) into the z components so the
  //    final 23-bit index feeds a uniform-base scale_offset b64 gather.
  const uint32_t lbase = l << kTBits;
  const uint32_t mx0 = g0;
  const uint32_t mx1 = g0 + 1u;
  const uint32_t hy0 = g1 * kPrime1;
  const uint32_t my0 = hy0 & kTMask;
  const uint32_t my1 = (hy0 + kPrime1) & kTMask;
  const uint32_t hz0 = g2 * kPrime2;
  const uint32_t mz0 = (hz0 & kTMask) | lbase;
  const uint32_t mz1 = ((hz0 + kPrime2) & kTMask) | lbase;

  // Corner v: bit0->dim0, bit1->dim1, bit2->dim2 (matches BORDER layout).
  const uint32_t i0 = mx0 ^ my0 ^ mz0;
  const uint32_t i1 = mx1 ^ my0 ^ mz0;
  const uint32_t i2 = mx0 ^ my1 ^ mz0;
  const uint32_t i3 = mx1 ^ my1 ^ mz0;
  const uint32_t i4 = mx0 ^ my0 ^ mz1;
  const uint32_t i5 = mx1 ^ my0 ^ mz1;
  const uint32_t i6 = mx0 ^ my1 ^ mz1;
  const uint32_t i7 = mx1 ^ my1 ^ mz1;

  // 8 gathers: global_load_b64 with SGPR base + 32-bit index, scale_offset.
  // Issued back-to-back so L2 latency overlaps (single load clause).
  const v2f* __restrict__ tb = (const v2f*)tables;
  const v2f t0 = tb[i0];
  const v2f t1 = tb[i1];
  const v2f t2 = tb[i2];
  const v2f t3 = tb[i3];
  const v2f t4 = tb[i4];
  const v2f t5 = tb[i5];
  const v2f t6 = tb[i6];
  const v2f t7 = tb[i7];

  // Trilinear weights exactly as reference: 1 - |s - (g + c)| per dim.
  const float wx0 = 1.0f - fabsf(s0 - f0);
  const float wx1 = 1.0f - fabsf(s0 - (f0 + 1.0f));
  const float wy0 = 1.0f - fabsf(s1 - f1);
  const float wy1 = 1.0f - fabsf(s1 - (f1 + 1.0f));
  const float wz0 = 1.0f - fabsf(s2 - f2);
  const float wz1 = 1.0f - fabsf(s2 - (f2 + 1.0f));

  const float p00 = wx0 * wy0;
  const float p10 = wx1 * wy0;
  const float p01 = wx0 * wy1;
  const float p11 = wx1 * wy1;

  const float w0 = p00 * wz0;
  const float w1 = p10 * wz0;
  const float w2 = p01 * wz0;
  const float w3 = p11 * wz0;
  const float w4 = p00 * wz1;
  const float w5 = p10 * wz1;
  const float w6 = p01 * wz1;
  const float w7 = p11 * wz1;

  float a0 = w0 * t0.x;
  float a1 = w0 * t0.y;
  a0 = fmaf(w1, t1.x, a0);  a1 = fmaf(w1, t1.y, a1);
  a0 = fmaf(w2, t2.x, a0);  a1 = fmaf(w2, t2.y, a1);
  a0 = fmaf(w3, t3.x, a0);  a1 = fmaf(w3, t3.y, a1);
  a0 = fmaf(w4, t4.x, a0);  a1 = fmaf(w4, t4.y, a1);
  a0 = fmaf(w5, t5.x, a0);  a1 = fmaf(w5, t5.y, a1);
  a0 = fmaf(w6, t6.x, a0);  a1 = fmaf(w6, t6.y, a1);
  a0 = fmaf(w7, t7.x, a0);  a1 = fmaf(w7, t7.y, a1);

  // Streamed 32MB output, never re-read: non-temporal store (TH=NT) so it
  // does not evict the L2-resident 64MB table working set. Consecutive
  // lanes -> consecutive 8B -> 256B contiguous per wave32.
  v2f o;
  o.x = a0;
  o.y = a1;
  __builtin_nontemporal_store(o, (v2f*)(out + (size_t)b * 32u + (size_t)l * 2u));
}

}  // namespace

extern "C" void kernel_launch(void* const* d_in, const int* in_sizes, int n_in,
                              void* d_out, int out_size, void* d_ws, size_t ws_size,
                              hipStream_t stream) {
  const float* x      = (const float*)d_in[0];   // (B, 3) f32
  const float* tables = (const float*)d_in[1];   // (16, 2^19, 2) f32
  float* out          = (float*)d_out;           // (B, 32) f32

  const int npts    = in_sizes[0] / 3;           // 262144
  const int threads = npts * kLevels;            // one thread per (point, level)
  const int block   = 256;                       // 8 wave32 per block
  const int grid    = (threads + block - 1) / block;

  ngp_interp_enc_kernel<<<grid, block, 0, stream>>>(x, tables, out, npts);
}